// VectorQuantizer_5042291605872
// MI455X (gfx1250) — compile-verified
//
#include <hip/hip_runtime.h>

typedef __attribute__((ext_vector_type(2))) float v2f;
typedef __attribute__((ext_vector_type(8))) float v8f;

#define Bn     16384
#define Dd     512
#define Nn     8
#define Kk     1024
#define KP     1025
#define Ee     64
#define DEPTHq 4
#define CB_STRIDE 68                        // 64 + 4 pad floats -> bank-conflict-free LDS rows
#define SMEM_FLOATS (Kk * CB_STRIDE + Kk)   // codebook + e_sq
#define SMEM_BYTES  (SMEM_FLOATS * 4)       // 282,624 B <= 320 KB WGP LDS

// ---------------------------------------------------------------------------
// Pass 1: zero the one-hot region (only scattered 1.0s are written later),
// and zero the loss / perplexity scalar slots.
// ---------------------------------------------------------------------------
__global__ void vq_fill_kernel(float* __restrict__ out,
                               float* __restrict__ onehot, long long n2) {
  long long i      = (long long)blockIdx.x * blockDim.x + threadIdx.x;
  long long stride = (long long)gridDim.x * blockDim.x;
  float2* p = (float2*)onehot;
  float2 z2 = make_float2(0.f, 0.f);
  for (long long j = i; j < n2; j += stride) p[j] = z2;
  if (i == 0) {
    out[0] = 0.f;                               // loss accumulator
    out[1 + (long long)Bn * Dd] = 0.f;          // perplexity = 0
  }
}

// ---------------------------------------------------------------------------
// Pass 2: residual VQ. grid = (32 row-tiles, 8 sub-spaces), block = 1024.
// Each wave32 owns a 16-row tile; codebook for this sub-space lives in LDS.
// ---------------------------------------------------------------------------
__global__ __launch_bounds__(1024)
void vq_rvq_kernel(const float* __restrict__ z,
                   const float* __restrict__ emb,
                   float* __restrict__ out) {
  extern __shared__ float smem[];             // [Kk*CB_STRIDE] codebook | [Kk] e_sq
  float* cb  = smem;
  float* esq = smem + Kk * CB_STRIDE;

  float* zq     = out + 1;
  float* onehot = out + 2 + (long long)Bn * Dd;
  float* idxOut = onehot + (long long)Bn * Nn * KP;

  const int tid = threadIdx.x;
  const int nId = blockIdx.y;

  // ---- stage codebook (1024 x 64 f32 = 256 KB) into LDS, fuse e_sq ----
  {
    const int code = tid;                     // 1024 threads -> 1 code row each
    const float4* src = (const float4*)(emb + ((size_t)nId * Kk + code) * Ee);
    float ss = 0.f;
#pragma unroll
    for (int q = 0; q < 16; ++q) {
      float4 v = src[q];
      ss += v.x * v.x + v.y * v.y + v.z * v.z + v.w * v.w;
      *(float4*)&cb[code * CB_STRIDE + q * 4] = v;    // 272B row stride, 16B aligned
    }
    esq[code] = ss;
  }
  __syncthreads();

  const int wave = tid >> 5;
  const int lane = tid & 31;
  const int m    = lane & 15;                 // my A-matrix row within the tile
  const int h    = lane >> 4;                 // K-half (f32 WMMA A layout)
  const int rowBase = blockIdx.x * 512 + wave * 16;
  const int row     = rowBase + m;

  // ---- load z tile in WMMA-A register layout: lane holds k = {4t+2h, 4t+2h+1} ----
  float zv[32], zr[32];
  {
    const float* zrow = z + (size_t)row * Dd + nId * Ee;
#pragma unroll
    for (int t = 0; t < 16; ++t) {
      float2 p = *(const float2*)(zrow + 4 * t + 2 * h);
      zv[2 * t] = p.x; zv[2 * t + 1] = p.y;
      zr[2 * t] = p.x; zr[2 * t + 1] = p.y;
    }
  }

  int lastIdx[8];
#pragma unroll 1
  for (int dep = 0; dep < DEPTHq; ++dep) {
    float best[8]; int bidx[8];
#pragma unroll
    for (int r = 0; r < 8; ++r) { best[r] = 0.f; bidx[r] = 0; }  // pad row: score 0, idx 0

#pragma unroll 1
    for (int chunk = 0; chunk < 64; ++chunk) {
      v8f acc = {};
      const int c = chunk * 16 + m;           // my B-matrix column's code
      const float* cbrow = &cb[c * CB_STRIDE + 2 * h];
#pragma unroll
      for (int t = 0; t < 16; ++t) {          // K = 64 in steps of 4
        v2f a = { zr[2 * t], zr[2 * t + 1] };
        float2 bb = *(const float2*)(cbrow + 4 * t);
        v2f b = { bb.x, bb.y };
        acc = __builtin_amdgcn_wmma_f32_16x16x4_f32(
            false, a, false, b, (short)0, acc, false, false);
      }
      const float es   = esq[c];
      const int   cand = c + 1;               // +1 for pad row at index 0
#pragma unroll
      for (int r = 0; r < 8; ++r) {
        float s = fmaf(-2.0f, acc[r], es);    // ||e||^2 - 2 z.e
        bool take = (s < best[r]) || (s == best[r] && cand < bidx[r]);
        if (take) { best[r] = s; bidx[r] = cand; }
      }
    }

    // ---- min-reduce (score, idx) across the 16 lanes of each half ----
#pragma unroll
    for (int off = 8; off >= 1; off >>= 1) {
#pragma unroll
      for (int r = 0; r < 8; ++r) {
        float os = __shfl_xor(best[r], off, 32);
        int   oi = __shfl_xor(bidx[r], off, 32);
        bool take = (os < best[r]) || (os == best[r] && oi < bidx[r]);
        if (take) { best[r] = os; bidx[r] = oi; }
      }
    }

    // ---- broadcast: lane needs idx of its own row m (rows 0-7 live in lanes
    // 0-15 regs 0-7, rows 8-15 in lanes 16-31) ----
    const int srcLane = (m >> 3) << 4;
    int myIdx = 0;
#pragma unroll
    for (int r = 0; r < 8; ++r) {
      int t = __shfl(bidx[r], srcLane, 32);
      if ((m & 7) == r) myIdx = t;
    }

    // ---- residual update from LDS codebook (idx 0 == pad == zero vector) ----
    if (myIdx > 0) {
      const float* q = &cb[(myIdx - 1) * CB_STRIDE + 2 * h];
#pragma unroll
      for (int t = 0; t < 16; ++t) {
        float2 p = *(const float2*)(q + 4 * t);
        zr[2 * t] -= p.x; zr[2 * t + 1] -= p.y;
      }
    }

    if (dep == DEPTHq - 1) {
#pragma unroll
      for (int r = 0; r < 8; ++r) lastIdx[r] = bidx[r];
    }
  }

  // ---- indices + one-hot scatter (last depth), from lanes 0 and 16 ----
  if ((lane & 15) == 0) {
#pragma unroll
    for (int r = 0; r < 8; ++r) {
      long long b  = rowBase + h * 8 + r;
      int       id = lastIdx[r];
      idxOut[b * Nn + nId] = (float)id;
      onehot[(b * Nn + nId) * KP + id] = 1.0f;
    }
  }

  // ---- z_q_st = z - z_res_final ; loss partial = 1.25 * mean(z_res^2) ----
  float ss = 0.f;
  {
    float* zqrow = zq + (size_t)row * Dd + nId * Ee;
#pragma unroll
    for (int t = 0; t < 16; ++t) {
      float a0 = zr[2 * t], a1 = zr[2 * t + 1];
      ss += a0 * a0 + a1 * a1;
      float2 o = make_float2(zv[2 * t] - a0, zv[2 * t + 1] - a1);
      *(float2*)(zqrow + 4 * t + 2 * h) = o;
    }
  }
#pragma unroll
  for (int off = 16; off >= 1; off >>= 1) ss += __shfl_xor(ss, off, 32);
  if (lane == 0)
    atomicAdd(out, ss * (1.25f / ((float)Bn * (float)Dd)));
}

// ---------------------------------------------------------------------------
extern "C" void kernel_launch(void* const* d_in, const int* in_sizes, int n_in,
                              void* d_out, int out_size, void* d_ws, size_t ws_size,
                              hipStream_t stream) {
  (void)in_sizes; (void)n_in; (void)d_ws; (void)ws_size; (void)out_size;
  const float* z   = (const float*)d_in[0];
  const float* emb = (const float*)d_in[1];
  float* out = (float*)d_out;

  float* onehot = out + 2 + (long long)Bn * Dd;
  long long n2 = ((long long)Bn * Nn * KP) / 2;      // float2 fill count

  vq_fill_kernel<<<8192, 256, 0, stream>>>(out, onehot, n2);

  dim3 grid(Bn / 512, Nn);                            // (32 row-tiles, 8 sub-spaces)
  vq_rvq_kernel<<<grid, 1024, SMEM_BYTES, stream>>>(z, emb, out);
}